// Quantizer_4939212390839
// MI455X (gfx1250) — compile-verified
//
#include <hip/hip_runtime.h>
#include <hip/hip_bf16.h>
#include <math.h>

typedef __attribute__((ext_vector_type(2))) float v2f;
typedef __attribute__((ext_vector_type(8))) float v8f;
typedef __attribute__((ext_vector_type(4))) unsigned int v4u;
typedef __attribute__((ext_vector_type(4))) int v4i;
typedef __attribute__((ext_vector_type(8))) int v8i;

#define NUM_EMBED 512
#define EMBED_DIM 64
#define SPATIAL   32768              // 32*32*32
#define BATCH     4
#define NTOK      (BATCH * SPATIAL)  // 131072 tokens
#define QELEMS    (BATCH * EMBED_DIM * SPATIAL)  // 8388608

// ---- output layout (floats, concatenated in reference return order) ----
#define OFF_LOSS      0
#define OFF_Q         1
#define OFF_PERP      (OFF_Q + QELEMS)
#define OFF_IDX       (OFF_PERP + 1)
#define OFF_NEWEMBED  (OFF_IDX + NTOK)
#define OFF_NEWCS     (OFF_NEWEMBED + NUM_EMBED * EMBED_DIM)
#define OFF_NEWEA     (OFF_NEWCS + NUM_EMBED)

// ---- workspace layout (floats) ----
#define WS_ENORM  0
#define WS_COUNT  (WS_ENORM + NUM_EMBED)
#define WS_DW     (WS_COUNT + NUM_EMBED)
#define WS_LOSS   (WS_DW + NUM_EMBED * EMBED_DIM)

// LDS codebook: row-major [code][LDS_PITCH floats]; pitch 66 = 64 + 2-DWORD pad.
// This is exactly what TDM pad_interval=5 (64 DWORDs) / pad_amount=1 (2 DWORDs)
// produces, keeps all 16 B-fragment ds_load_b64s within DS immediate-offset
// range of one base address, and gives bank-stride-2 conflict-free reads.
#define LDS_PITCH 66
#define LDS_BYTES (NUM_EMBED * LDS_PITCH * 4)   // 135168 B (< 320 KB/WGP)

#if __has_builtin(__builtin_amdgcn_tensor_load_to_lds)
#define HAVE_TDM 1
#if __has_include(<hip/amd_detail/amd_gfx1250_TDM.h>)
#define TDM_ARITY6 1   // therock headers => 6-arg builtin
#endif
#endif

// ---------------------------------------------------------------------------
// Kernel 1: precompute ||e_k||^2, zero accumulators
// ---------------------------------------------------------------------------
__global__ void vq_init(const float* __restrict__ embed, float* __restrict__ ws) {
    int t = threadIdx.x;  // 512 threads
    float s = 0.f;
    for (int c = 0; c < EMBED_DIM; ++c) {
        float v = embed[t * EMBED_DIM + c];
        s += v * v;
    }
    ws[WS_ENORM + t] = s;
    ws[WS_COUNT + t] = 0.f;
    for (int i = t; i < NUM_EMBED * EMBED_DIM; i += 512) ws[WS_DW + i] = 0.f;
    if (t == 0) ws[WS_LOSS] = 0.f;
}

// ---------------------------------------------------------------------------
// Kernel 2: WMMA distance GEMM + argmin + gather/scatter
//   A (16 x 64) = token rows, B (64 x 512) = embed^T, accumulate f32 via
//   V_WMMA_F32_16X16X4_F32. dist = ||e||^2 - 2*(x.e)  (||x||^2 is row-const)
// ---------------------------------------------------------------------------
__global__ void __launch_bounds__(256, 1)
vq_wmma(const float* __restrict__ x, const float* __restrict__ embed,
        float* __restrict__ out, float* __restrict__ ws) {
    extern __shared__ float ldsE[];         // [NUM_EMBED][LDS_PITCH]
    __shared__ int s_idx[8 * 16];           // per-wave winning code per row

    const int tid  = threadIdx.x;
    const int lane = tid & 31;
    const int wave = tid >> 5;
    const int ln   = lane & 15;
    const int half = lane >> 4;

    // ---- stage codebook into LDS (row-major, pitch 66)
#if HAVE_TDM
    if (wave == 0) {
        // Tensor DMA descriptor (D#) per CDNA5 ISA §8:
        unsigned lds_base = (unsigned)(size_t)(void*)ldsE;
        unsigned long long ga = (unsigned long long)(size_t)embed;
        v4u g0;
        g0[0] = 1u;                                   // count=1 (valid), load, no gather
        g0[1] = lds_base;                             // lds_addr
        g0[2] = (unsigned)(ga & 0xFFFFFFFFu);         // global_addr[31:0]
        g0[3] = (unsigned)((ga >> 32) & 0x01FFFFFFu)  // global_addr[56:32]
              | (2u << 30);                           // type=2 ("image")
        v8i g1;
        g1[0] = (int)((2u << 16)      // data_size = 4 bytes
                    | (1u << 20)      // pad_enable
                    | (5u << 22)      // pad_interval: every 64 DWORDs
                    | (1u << 25));    // pad_amount: 2 DWORDs  -> pitch 66
        g1[1] = (int)(64u << 16);     // tensor_dim0 = 64  (bits [79:48])
        g1[2] = (int)(512u << 16);    // tensor_dim1 = 512 (bits [111:80])
        g1[3] = (int)(64u << 16);     // tile_dim0 = 64    (bits [127:112])
        g1[4] = 512;                  // tile_dim1 = 512
        g1[5] = 64;                   // tensor_dim0_stride = 64
        g1[6] = 0;
        g1[7] = 0;
        v4i gz = {0, 0, 0, 0};
#if TDM_ARITY6
        v8i gz8 = {0, 0, 0, 0, 0, 0, 0, 0};
        __builtin_amdgcn_tensor_load_to_lds(g0, g1, gz, gz, gz8, 0);
#else
        __builtin_amdgcn_tensor_load_to_lds(g0, g1, gz, gz, 0);
#endif
        __builtin_amdgcn_s_wait_tensorcnt(0);
    }
#else
    {
        const float2* e2 = (const float2*)embed;      // [512][32] float2
        for (int e = tid; e < NUM_EMBED * 32; e += 256) {
            int code = e >> 5, cp = e & 31;
            *(float2*)&ldsE[code * LDS_PITCH + 2 * cp] = e2[code * 32 + cp];
        }
    }
#endif
    __syncthreads();

    // ---- this wave's 16-row token block
    const int rowBase = (blockIdx.x * 8 + wave) * 16;
    const int b   = rowBase >> 15;            // batch index (SPATIAL = 2^15)
    const int sp0 = rowBase & (SPATIAL - 1);  // block never crosses batch
    const float* xblk = x + (size_t)b * EMBED_DIM * SPATIAL + sp0;

    // ---- preload A fragment (16x4 f32 layout: lane half -> K offset 0/2)
    const int m   = ln;
    const int off = half * 2;
    float aReg[32];
#pragma unroll
    for (int kk = 0; kk < 16; ++kk) {
        int c0 = 4 * kk + off;
        aReg[2 * kk]     = xblk[(size_t)c0 * SPATIAL + m];
        aReg[2 * kk + 1] = xblk[(size_t)(c0 + 1) * SPATIAL + m];
    }

    float bestD[8];
    int   bestI[8];
#pragma unroll
    for (int r = 0; r < 8; ++r) { bestD[r] = 3.4e38f; bestI[r] = 0; }

    const float* enorm = ws + WS_ENORM;

    // ---- sweep 32 column tiles of 16 codes each
    for (int kt = 0; kt < 32; ++kt) {
        const float* brow = &ldsE[(kt * 16 + ln) * LDS_PITCH + off];
        v8f acc = {};
#pragma unroll
        for (int kk = 0; kk < 16; ++kk) {
            // B fragment: K = 4kk+off(+1), N = kt*16+ln; immediate ds offsets
            float2 bp = *(const float2*)&brow[4 * kk];
            v2f a;  a[0] = aReg[2 * kk];  a[1] = aReg[2 * kk + 1];
            v2f bb; bb[0] = bp.x;         bb[1] = bp.y;
            acc = __builtin_amdgcn_wmma_f32_16x16x4_f32(
                false, a, false, bb, (short)0, acc, false, false);
        }
        int   code = kt * 16 + ln;
        float en   = enorm[code];
#pragma unroll
        for (int r = 0; r < 8; ++r) {
            float d = en - 2.0f * acc[r];
            if (d < bestD[r]) { bestD[r] = d; bestI[r] = code; }
        }
    }

    // ---- cross-lane argmin within each 16-lane half (N is striped over lanes)
#pragma unroll
    for (int mask = 1; mask <= 8; mask <<= 1) {
#pragma unroll
        for (int r = 0; r < 8; ++r) {
            float od = __shfl_xor(bestD[r], mask, 32);
            int   oi = __shfl_xor(bestI[r], mask, 32);
            if (od < bestD[r] || (od == bestD[r] && oi < bestI[r])) {
                bestD[r] = od; bestI[r] = oi;
            }
        }
    }
    // lane 0 holds rows m=r (half 0); lane 16 holds rows m=r+8 (half 1)
    if (ln == 0) {
#pragma unroll
        for (int r = 0; r < 8; ++r) s_idx[wave * 16 + half * 8 + r] = bestI[r];
    }
    __builtin_amdgcn_wave_barrier();   // wave-coherent LDS handoff

    // ---- gather + quantized store + loss + EMA segment-sum accumulation
    float  lossAcc = 0.f;
    float* outQ   = out + OFF_Q;
    float* outIdx = out + OFF_IDX;
    float* dw     = ws + WS_DW;
    float* cnt    = ws + WS_COUNT;
    const int cp  = lane;              // channel pair owned by this lane (0..31)

    for (int mm = 0; mm < 16; ++mm) {
        int idx = s_idx[wave * 16 + mm];
        int row = rowBase + mm;
        float2 q = *(const float2*)&ldsE[idx * LDS_PITCH + 2 * cp];
        size_t base0 = (size_t)(b * EMBED_DIM + 2 * cp) * SPATIAL + (sp0 + mm);
        size_t base1 = base0 + SPATIAL;
        float x0 = x[base0], x1 = x[base1];
        outQ[base0] = q.x;                          // straight-through: out == quantized
        outQ[base1] = q.y;
        float d0 = q.x - x0, d1 = q.y - x1;
        lossAcc += d0 * d0 + d1 * d1;
        atomicAdd(&dw[idx * EMBED_DIM + 2 * cp],     x0);
        atomicAdd(&dw[idx * EMBED_DIM + 2 * cp + 1], x1);
        if (lane == 0) {
            outIdx[row] = (float)idx;
            atomicAdd(&cnt[idx], 1.0f);
        }
    }

    // ---- loss: full-wave reduce, single atomic per wave
#pragma unroll
    for (int mask = 1; mask < 32; mask <<= 1) lossAcc += __shfl_xor(lossAcc, mask, 32);
    if (lane == 0) atomicAdd(&ws[WS_LOSS], lossAcc);
}

// ---------------------------------------------------------------------------
// Kernel 3: EMA update, Laplace smoothing, perplexity, loss scalar
// ---------------------------------------------------------------------------
__global__ void vq_finalize(const float* __restrict__ embed_avg,
                            const float* __restrict__ cluster_size,
                            float* __restrict__ out, const float* __restrict__ ws) {
    __shared__ float red[NUM_EMBED];
    int k = threadIdx.x;  // 512 threads
    float cntk = ws[WS_COUNT + k];
    float newc = cluster_size[k] * 0.99f + 0.01f * cntk;
    out[OFF_NEWCS + k] = newc;

    red[k] = newc;
    __syncthreads();
    for (int s = 256; s > 0; s >>= 1) {
        if (k < s) red[k] += red[k + s];
        __syncthreads();
    }
    float n = red[0];
    __syncthreads();

    float avgp = cntk / (float)NTOK;
    red[k] = avgp * logf(avgp + 1e-10f);
    __syncthreads();
    for (int s = 256; s > 0; s >>= 1) {
        if (k < s) red[k] += red[k + s];
        __syncthreads();
    }
    if (k == 0) {
        out[OFF_PERP] = expf(-red[0]);
        out[OFF_LOSS] = 0.25f * ws[WS_LOSS] / (float)QELEMS;
    }

    float smoothed = n * ((newc + 1e-5f) / (n + NUM_EMBED * 1e-5f));
    float inv = 1.0f / smoothed;
    for (int c = 0; c < EMBED_DIM; ++c) {
        float nea = embed_avg[k * EMBED_DIM + c] * 0.99f
                  + 0.01f * ws[WS_DW + k * EMBED_DIM + c];
        out[OFF_NEWEA + k * EMBED_DIM + c]     = nea;
        out[OFF_NEWEMBED + k * EMBED_DIM + c]  = nea * inv;
    }
}

// ---------------------------------------------------------------------------
extern "C" void kernel_launch(void* const* d_in, const int* in_sizes, int n_in,
                              void* d_out, int out_size, void* d_ws, size_t ws_size,
                              hipStream_t stream) {
    const float* x            = (const float*)d_in[0];  // [4,64,32,32,32]
    const float* embed        = (const float*)d_in[1];  // [512,64]
    const float* embed_avg    = (const float*)d_in[2];  // [512,64]
    const float* cluster_size = (const float*)d_in[3];  // [512]
    float* out = (float*)d_out;
    float* ws  = (float*)d_ws;

    // allow >64KB dynamic LDS (MI455X WGP has 320KB)
    (void)hipFuncSetAttribute(reinterpret_cast<const void*>(vq_wmma),
                              hipFuncAttributeMaxDynamicSharedMemorySize,
                              (int)LDS_BYTES);

    vq_init<<<1, NUM_EMBED, 0, stream>>>(embed, ws);
    vq_wmma<<<NTOK / 128, 256, LDS_BYTES, stream>>>(x, embed, out, ws);
    vq_finalize<<<1, NUM_EMBED, 0, stream>>>(embed_avg, cluster_size, out, ws);
}